// Model_Regressor_16406775071385
// MI455X (gfx1250) — compile-verified
//
#include <hip/hip_runtime.h>
#include <math.h>

#define NN       50000
#define NE       1600000
#define ETOT     (NE + NN)        // edges + self loops
#define NF       256
#define HD       64
#define NHEADS   8
#define NGRAPHS  64
#define NEG_SLOPE 0.2f

typedef float v2f __attribute__((ext_vector_type(2)));
typedef float v8f __attribute__((ext_vector_type(8)));

// ---- workspace layout (offsets in floats) ----
#define OFF_H1    0L         // [NN*64]
#define OFF_AS1   3200000L   // [NN*8]
#define OFF_AD1   3600000L   // [NN*8]
#define OFF_M1    4000000L   // [NN*8]  (ordered-uint encoded max)
#define OFF_DEN1  4400000L   // [NN*8]
#define OFF_AGG1  4800000L   // [NN*64]
#define OFF_H2    8000000L   // [NN]
#define OFF_AS2   8050000L   // [NN]
#define OFF_AD2   8100000L   // [NN]
#define OFF_M2    8150000L   // [NN]    (ordered-uint encoded max)
#define OFF_DEN2  8200000L   // [NN]
#define OFF_AGG2  8250000L   // [NN]
#define OFF_GSUM  8300000L   // [64]
#define OFF_GCNT  8300064L   // [64]
#define WS_FLOATS 8300128L

// order-preserving float<->uint encoding: enc is monotone, enc=0 < enc(any float)
__device__ __forceinline__ unsigned enc_ord(float f) {
  unsigned u = __float_as_uint(f);
  return (u & 0x80000000u) ? ~u : (u | 0x80000000u);
}
__device__ __forceinline__ float dec_ord(unsigned u) {
  return (u & 0x80000000u) ? __uint_as_float(u & 0x7fffffffu) : __uint_as_float(~u);
}
__device__ __forceinline__ float lrelu(float x) { return x > 0.f ? x : NEG_SLOPE * x; }

__device__ __forceinline__ void edge_sd(const int* __restrict__ ei, int e, int& s, int& d) {
  if (e < NE) { s = ei[e]; d = ei[NE + e]; } else { s = e - NE; d = e - NE; }
}

// -------------------- init --------------------
__global__ void zero_kernel(float* __restrict__ p, long n) {
  long i = (long)blockIdx.x * blockDim.x + threadIdx.x;
  if (i < n) p[i] = 0.f;
}

// -------------------- layer 1 GEMM: h1 = x @ W1 (fp32 WMMA) --------------------
// B fragments are pre-swizzled into LDS so each lane's v2f B-operand is one
// conflict-free ds_load_b64 (no cross-register repacking in the inner loop).
__global__ __launch_bounds__(256) void gemm1_kernel(const float* __restrict__ x,
                                                    const float* __restrict__ W,
                                                    float* __restrict__ h1) {
  // lb[(kstep*4 + nt)*32 + lane] = { W[k][col], W[k+1][col] }
  //   with k = kstep*4 + 2*(lane>>4), col = nt*16 + (lane&15)
  __shared__ v2f lb[(NF / 4) * 4 * 32];   // 8192 v2f = 64 KB
  for (int idx = threadIdx.x; idx < (NF / 4) * 4 * 32; idx += 256) {
    int kstep = idx >> 7;
    int nt    = (idx >> 5) & 3;
    int ln    = idx & 31;
    int k     = kstep * 4 + ((ln >> 4) << 1);
    int col   = nt * 16 + (ln & 15);
    v2f f; f.x = W[k * HD + col]; f.y = W[(k + 1) * HD + col];
    lb[idx] = f;
  }
  __syncthreads();

  const int wave = threadIdx.x >> 5;
  const int lane = threadIdx.x & 31;
  const int tile = blockIdx.x * 8 + wave;       // 16-row M tile per wave; 50000/16 = 3125 exact
  if (tile >= NN / 16) return;

  const int ncol  = lane & 15;
  const int khalf = (lane >> 4) << 1;           // lanes 0-15: K=0,1 ; lanes 16-31: K=2,3
  const int row   = tile * 16 + (lane & 15);
  const float* __restrict__ xr = x + (long)row * NF;

  v8f acc0 = {}, acc1 = {}, acc2 = {}, acc3 = {};
  for (int kstep = 0; kstep < NF / 4; kstep++) {
    v2f a;
    a.x = xr[kstep * 4 + khalf];
    a.y = xr[kstep * 4 + khalf + 1];
    const v2f* bb = lb + kstep * 128 + lane;    // +0, +32, +64, +96 -> 4 n-tiles
    v2f b0 = bb[0];
    v2f b1 = bb[32];
    v2f b2 = bb[64];
    v2f b3 = bb[96];
    acc0 = __builtin_amdgcn_wmma_f32_16x16x4_f32(false, a, false, b0, (short)0, acc0, false, false);
    acc1 = __builtin_amdgcn_wmma_f32_16x16x4_f32(false, a, false, b1, (short)0, acc1, false, false);
    acc2 = __builtin_amdgcn_wmma_f32_16x16x4_f32(false, a, false, b2, (short)0, acc2, false, false);
    acc3 = __builtin_amdgcn_wmma_f32_16x16x4_f32(false, a, false, b3, (short)0, acc3, false, false);
  }

  // D layout: VGPR r holds M = r (lanes 0-15) or M = 8+r (lanes 16-31), N = lane&15
  const int mbase = tile * 16 + ((lane >> 4) << 3);
  #pragma unroll
  for (int r = 0; r < 8; r++) {
    long o = (long)(mbase + r) * HD + ncol;
    h1[o]      = acc0[r];
    h1[o + 16] = acc1[r];
    h1[o + 32] = acc2[r];
    h1[o + 48] = acc3[r];
  }
}

// -------------------- per-node attention logits (layer 1) --------------------
__global__ void alpha1_kernel(const float* __restrict__ h1,
                              const float* __restrict__ asrc,
                              const float* __restrict__ adst,
                              float* __restrict__ as1, float* __restrict__ ad1) {
  int n = blockIdx.x * blockDim.x + threadIdx.x;
  if (n >= NN) return;
  const float* r = h1 + (long)n * HD;
  #pragma unroll
  for (int h = 0; h < NHEADS; h++) {
    float s = 0.f, d = 0.f;
    #pragma unroll
    for (int c = 0; c < 8; c++) {
      float v = r[h * 8 + c];
      s += v * asrc[h * 8 + c];
      d += v * adst[h * 8 + c];
    }
    as1[n * NHEADS + h] = s;
    ad1[n * NHEADS + h] = d;
  }
}

// -------------------- layer 1 edge passes --------------------
__global__ void edge_max1_kernel(const int* __restrict__ ei,
                                 const float* __restrict__ as1, const float* __restrict__ ad1,
                                 unsigned* __restrict__ m1) {
  long t = (long)blockIdx.x * blockDim.x + threadIdx.x;
  if (t >= (long)ETOT * NHEADS) return;
  int e = (int)(t >> 3), h = (int)(t & 7);
  int s, d; edge_sd(ei, e, s, d);
  float ev = lrelu(as1[s * NHEADS + h] + ad1[d * NHEADS + h]);
  atomicMax(&m1[d * NHEADS + h], enc_ord(ev));
}

__global__ void edge_den1_kernel(const int* __restrict__ ei,
                                 const float* __restrict__ as1, const float* __restrict__ ad1,
                                 const unsigned* __restrict__ m1, float* __restrict__ den1) {
  long t = (long)blockIdx.x * blockDim.x + threadIdx.x;
  if (t >= (long)ETOT * NHEADS) return;
  int e = (int)(t >> 3), h = (int)(t & 7);
  int s, d; edge_sd(ei, e, s, d);
  float ev = lrelu(as1[s * NHEADS + h] + ad1[d * NHEADS + h]);
  float p = expf(ev - dec_ord(m1[d * NHEADS + h]));
  atomicAdd(&den1[d * NHEADS + h], p);
}

__global__ void edge_agg1_kernel(const int* __restrict__ ei,
                                 const float* __restrict__ as1, const float* __restrict__ ad1,
                                 const unsigned* __restrict__ m1, const float* __restrict__ den1,
                                 const float* __restrict__ h1, float* __restrict__ agg1) {
  long t = (long)blockIdx.x * blockDim.x + threadIdx.x;
  if (t >= (long)ETOT * NHEADS) return;
  int e = (int)(t >> 3), h = (int)(t & 7);
  int s, d; edge_sd(ei, e, s, d);
  float ev = lrelu(as1[s * NHEADS + h] + ad1[d * NHEADS + h]);
  float p = expf(ev - dec_ord(m1[d * NHEADS + h]));
  float w = p / den1[d * NHEADS + h];
  const float* hs = h1 + (long)s * HD + h * 8;
  float* od = agg1 + (long)d * HD + h * 8;
  #pragma unroll
  for (int c = 0; c < 8; c++) atomicAdd(od + c, w * hs[c]);
}

// -------------------- inter-layer: bias+relu, h2 = relu(out1) @ W2, layer-2 logits --------------------
__global__ void node2_kernel(const float* __restrict__ agg1, const float* __restrict__ b1,
                             const float* __restrict__ W2,
                             const float* __restrict__ asrc2, const float* __restrict__ adst2,
                             float* __restrict__ h2, float* __restrict__ as2, float* __restrict__ ad2) {
  int n = blockIdx.x * blockDim.x + threadIdx.x;
  if (n >= NN) return;
  const float* r = agg1 + (long)n * HD;
  float acc = 0.f;
  #pragma unroll
  for (int i = 0; i < HD; i++) {
    float v = fmaxf(r[i] + b1[i], 0.f);
    acc += v * W2[i];
  }
  h2[n] = acc;
  as2[n] = acc * asrc2[0];
  ad2[n] = acc * adst2[0];
}

// -------------------- layer 2 edge passes (single head, single channel) --------------------
__global__ void edge_max2_kernel(const int* __restrict__ ei,
                                 const float* __restrict__ as2, const float* __restrict__ ad2,
                                 unsigned* __restrict__ m2) {
  int e = blockIdx.x * blockDim.x + threadIdx.x;
  if (e >= ETOT) return;
  int s, d; edge_sd(ei, e, s, d);
  atomicMax(&m2[d], enc_ord(lrelu(as2[s] + ad2[d])));
}

__global__ void edge_den2_kernel(const int* __restrict__ ei,
                                 const float* __restrict__ as2, const float* __restrict__ ad2,
                                 const unsigned* __restrict__ m2, float* __restrict__ den2) {
  int e = blockIdx.x * blockDim.x + threadIdx.x;
  if (e >= ETOT) return;
  int s, d; edge_sd(ei, e, s, d);
  float p = expf(lrelu(as2[s] + ad2[d]) - dec_ord(m2[d]));
  atomicAdd(&den2[d], p);
}

__global__ void edge_agg2_kernel(const int* __restrict__ ei,
                                 const float* __restrict__ as2, const float* __restrict__ ad2,
                                 const unsigned* __restrict__ m2, const float* __restrict__ den2,
                                 const float* __restrict__ h2, float* __restrict__ agg2) {
  int e = blockIdx.x * blockDim.x + threadIdx.x;
  if (e >= ETOT) return;
  int s, d; edge_sd(ei, e, s, d);
  float p = expf(lrelu(as2[s] + ad2[d]) - dec_ord(m2[d]));
  float w = p / den2[d];
  atomicAdd(&agg2[d], w * h2[s]);
}

// -------------------- global mean pool --------------------
__global__ void pool_kernel(const float* __restrict__ agg2, const float* __restrict__ b2,
                            const int* __restrict__ batch,
                            float* __restrict__ gsum, float* __restrict__ gcnt) {
  int n = blockIdx.x * blockDim.x + threadIdx.x;
  if (n >= NN) return;
  float v = agg2[n] + b2[0];
  int g = batch[n];
  atomicAdd(&gsum[g], v);
  atomicAdd(&gcnt[g], 1.0f);
}

__global__ void final_kernel(const float* __restrict__ gsum, const float* __restrict__ gcnt,
                             float* __restrict__ out) {
  int g = threadIdx.x;
  if (g < NGRAPHS) out[g] = gsum[g] / fmaxf(gcnt[g], 1.0f);
}

// -------------------- launcher --------------------
extern "C" void kernel_launch(void* const* d_in, const int* in_sizes, int n_in,
                              void* d_out, int out_size, void* d_ws, size_t ws_size,
                              hipStream_t stream) {
  const float* x     = (const float*)d_in[0];
  const int*   ei    = (const int*)d_in[1];
  const int*   batch = (const int*)d_in[2];
  const float* W1    = (const float*)d_in[3];
  const float* asrc1 = (const float*)d_in[4];
  const float* adst1 = (const float*)d_in[5];
  const float* b1    = (const float*)d_in[6];
  const float* W2    = (const float*)d_in[7];
  const float* asrc2 = (const float*)d_in[8];
  const float* adst2 = (const float*)d_in[9];
  const float* b2    = (const float*)d_in[10];
  float* ws  = (float*)d_ws;
  float* out = (float*)d_out;

  // zero all accumulators / max-state (ordered-uint 0 == "-inf")
  long nz = WS_FLOATS - OFF_M1;
  zero_kernel<<<(int)((nz + 255) / 256), 256, 0, stream>>>(ws + OFF_M1, nz);

  // layer 1: GEMM + logits
  gemm1_kernel<<<(NN / 16 + 7) / 8, 256, 0, stream>>>(x, W1, ws + OFF_H1);
  alpha1_kernel<<<(NN + 255) / 256, 256, 0, stream>>>(ws + OFF_H1, asrc1, adst1,
                                                      ws + OFF_AS1, ws + OFF_AD1);
  // layer 1 edge passes
  long eth = (long)ETOT * NHEADS;
  int  gEH = (int)((eth + 255) / 256);
  edge_max1_kernel<<<gEH, 256, 0, stream>>>(ei, ws + OFF_AS1, ws + OFF_AD1,
                                            (unsigned*)(ws + OFF_M1));
  edge_den1_kernel<<<gEH, 256, 0, stream>>>(ei, ws + OFF_AS1, ws + OFF_AD1,
                                            (const unsigned*)(ws + OFF_M1), ws + OFF_DEN1);
  edge_agg1_kernel<<<gEH, 256, 0, stream>>>(ei, ws + OFF_AS1, ws + OFF_AD1,
                                            (const unsigned*)(ws + OFF_M1), ws + OFF_DEN1,
                                            ws + OFF_H1, ws + OFF_AGG1);
  // inter-layer
  node2_kernel<<<(NN + 255) / 256, 256, 0, stream>>>(ws + OFF_AGG1, b1, W2, asrc2, adst2,
                                                     ws + OFF_H2, ws + OFF_AS2, ws + OFF_AD2);
  // layer 2 edge passes
  int gE = (ETOT + 255) / 256;
  edge_max2_kernel<<<gE, 256, 0, stream>>>(ei, ws + OFF_AS2, ws + OFF_AD2,
                                           (unsigned*)(ws + OFF_M2));
  edge_den2_kernel<<<gE, 256, 0, stream>>>(ei, ws + OFF_AS2, ws + OFF_AD2,
                                           (const unsigned*)(ws + OFF_M2), ws + OFF_DEN2);
  edge_agg2_kernel<<<gE, 256, 0, stream>>>(ei, ws + OFF_AS2, ws + OFF_AD2,
                                           (const unsigned*)(ws + OFF_M2), ws + OFF_DEN2,
                                           ws + OFF_H2, ws + OFF_AGG2);
  // pool
  pool_kernel<<<(NN + 255) / 256, 256, 0, stream>>>(ws + OFF_AGG2, b2, batch,
                                                    ws + OFF_GSUM, ws + OFF_GCNT);
  final_kernel<<<1, 64, 0, stream>>>(ws + OFF_GSUM, ws + OFF_GCNT, out);
}